// Superpoint_MAE_67585605369902
// MI455X (gfx1250) — compile-verified
//
#include <hip/hip_runtime.h>
#include <hip/hip_bf16.h>

// ---------------------------------------------------------------------------
// Superpoint MAE token embed:  relu(relu(X@W1+b1)@W2+b2) -> segment_max
//   X: [1.5M, 32] f32, W1: [32,64], b1:[64], W2:[64,128], b2:[128]
//   out: [50000, 128] f32 (empty segments -> 0)
//
// MI455X / gfx1250 (wave32) strategy:
//   * fp32 WMMA (v_wmma_f32_16x16x4_f32) -> bit-exact vs fp32 reference
//   * each wave processes TWO 16-point tiles per iteration so every LDS
//     B-fragment load feeds two WMMAs (halves DS issue per point)
//   * weights stored TRANSPOSED in LDS with padded strides (36 / 68) so a
//     B fragment is a single aligned ds_load_b64 and the bank pattern
//     (4*lane + k) mod 64 is conflict-free across all 32 lanes
//   * layer-1 C/D layout -> A layout conversion through wave-private LDS
//     tiles (row stride 68 -> conflict-free), ordered by s_wait_dscnt
//   * segment max via global_atomic_max_u32 on float bits (post-ReLU values
//     are >= 0, out is zero-initialized -> exactly matches the reference
//     including the empty-segment -> 0 guard)
// ---------------------------------------------------------------------------

typedef __attribute__((ext_vector_type(2))) float v2f;
typedef __attribute__((ext_vector_type(8))) float v8f;

namespace spcfg {
constexpr int N_POINTS = 1500000;
constexpr int FEAT = 32;
constexpr int HID  = 64;
constexpr int EMB  = 128;
constexpr int NPAIRS = N_POINTS / 32;      // 46875 (exact: 32 points / pair)
constexpr int BLOCK = 64;                  // 2 waves of 32
constexpr int WAVES_PER_BLOCK = BLOCK / 32;
constexpr int GRID = 2500;                 // 5000 waves total
constexpr int W1S = FEAT + 4;              // 36: padded k-stride of W1^T
constexpr int W2S = HID + 4;               // 68: padded k-stride of W2^T
constexpr int HS  = HID + 4;               // 68: padded row stride of h1 tile
}

__global__ void Superpoint_MAE_zero_kernel(float4* __restrict__ out, int n4) {
    int i = blockIdx.x * blockDim.x + threadIdx.x;
    if (i < n4) out[i] = make_float4(0.f, 0.f, 0.f, 0.f);
}

__launch_bounds__(spcfg::BLOCK)
__global__ void Superpoint_MAE_mlp_segmax_kernel(
    const float* __restrict__ X,
    const int*   __restrict__ seg,
    const float* __restrict__ W1,
    const float* __restrict__ b1,
    const float* __restrict__ W2,
    const float* __restrict__ b2,
    unsigned int* __restrict__ out)
{
    using namespace spcfg;

    __shared__ float sW1t[HID * W1S];                 //  9216 B, [n][k]
    __shared__ float sW2t[EMB * W2S];                 // 34816 B, [n][k]
    __shared__ float sB1[HID];
    __shared__ float sB2[EMB];
    __shared__ float sH[2 * WAVES_PER_BLOCK][16 * HS]; // 17408 B

    // ---- stage transposed weights/biases into LDS (once per block) ----
    for (int i = threadIdx.x; i < FEAT * HID; i += BLOCK) {
        int k = i >> 6, n = i & (HID - 1);            // W1[k][n], contiguous read
        sW1t[n * W1S + k] = W1[i];
    }
    for (int i = threadIdx.x; i < HID * EMB; i += BLOCK) {
        int k = i >> 7, n = i & (EMB - 1);            // W2[k][n], contiguous read
        sW2t[n * W2S + k] = W2[i];
    }
    if (threadIdx.x < HID) sB1[threadIdx.x] = b1[threadIdx.x];
    if (threadIdx.x < EMB) sB2[threadIdx.x] = b2[threadIdx.x];
    __syncthreads();

    const int wave = threadIdx.x >> 5;
    const int lane = threadIdx.x & 31;
    const int nl   = lane & 15;       // N / M position within half-wave
    const int h    = lane >> 4;       // half-wave select
    const int h2   = h * 2;           // K offset of this half (f32 WMMA layout)
    float* myH0 = &sH[wave * 2 + 0][0];
    float* myH1 = &sH[wave * 2 + 1][0];

    const int wavesTotal = gridDim.x * WAVES_PER_BLOCK;

    for (int pair = blockIdx.x * WAVES_PER_BLOCK + wave; pair < NPAIRS;
         pair += wavesTotal) {
        const int base = pair * 32;

        // prefetch next pair's X rows into cache (global_prefetch_b8 path)
        if (pair + wavesTotal < NPAIRS)
            __builtin_prefetch(X + (long)(base + wavesTotal * 32) * FEAT +
                                   lane * FEAT, 0, 1);

        // ---- A fragments for layer 1: two X tiles [16 x 32] ----
        // A 16x4 f32 layout: lane = row M (0..15), VGPR v & half h -> K = 2h+v
        v2f a1[2][8];
#pragma unroll
        for (int u = 0; u < 2; ++u) {
            const float* xrow = X + (long)(base + u * 16 + nl) * FEAT;
#pragma unroll
            for (int kk = 0; kk < 8; ++kk)
                a1[u][kk] = *(const v2f*)(xrow + kk * 4 + h2);
        }

        // ---- layer 1: [16x32] @ [32x64], 4 N-tiles x 8 K-steps x 2 tiles ----
        v8f acc1[2][4];
#pragma unroll
        for (int t = 0; t < 4; ++t) {
            const float bv = sB1[t * 16 + nl];
#pragma unroll
            for (int r = 0; r < 8; ++r) { acc1[0][t][r] = bv; acc1[1][t][r] = bv; }
#pragma unroll
            for (int kk = 0; kk < 8; ++kk) {
                // B 4x16 f32 layout: lane = col N, VGPR v & half h -> K = 2h+v
                // single aligned ds_load_b64 from transposed, padded W1^T
                const v2f bf =
                    *(const v2f*)&sW1t[(t * 16 + nl) * W1S + kk * 4 + h2];
                acc1[0][t] = __builtin_amdgcn_wmma_f32_16x16x4_f32(
                    false, a1[0][kk], false, bf, (short)0, acc1[0][t], false, false);
                acc1[1][t] = __builtin_amdgcn_wmma_f32_16x16x4_f32(
                    false, a1[1][kk], false, bf, (short)0, acc1[1][t], false, false);
            }
#pragma unroll
            for (int r = 0; r < 8; ++r) {
                acc1[0][t][r] = fmaxf(acc1[0][t][r], 0.0f);
                acc1[1][t][r] = fmaxf(acc1[1][t][r], 0.0f);
            }
        }

        // ---- C/D layout -> row-major [16][HS] in wave-private LDS ----
        // C/D: VGPR r & half h -> row M = r + 8h ; lane nl -> col t*16+nl
#pragma unroll
        for (int t = 0; t < 4; ++t)
#pragma unroll
            for (int r = 0; r < 8; ++r) {
                myH0[(r + h * 8) * HS + t * 16 + nl] = acc1[0][t][r];
                myH1[(r + h * 8) * HS + t * 16 + nl] = acc1[1][t][r];
            }
        // DS ops are in-order within a wave; pin ordering + wait explicitly.
        asm volatile("s_wait_dscnt 0" ::: "memory");

        // ---- A fragments for layer 2: two h1 tiles [16 x 64] ----
        v2f a2[2][16];
#pragma unroll
        for (int kk = 0; kk < 16; ++kk) {
            a2[0][kk] = *(const v2f*)(myH0 + nl * HS + kk * 4 + h2);
            a2[1][kk] = *(const v2f*)(myH1 + nl * HS + kk * 4 + h2);
        }

        // segment ids for the 8 rows this half-wave owns in C/D layout
        int sidx[2][8];
#pragma unroll
        for (int u = 0; u < 2; ++u)
#pragma unroll
            for (int r = 0; r < 8; ++r)
                sidx[u][r] = seg[base + u * 16 + r + h * 8];

        // ---- layer 2: [16x64] @ [64x128], 8 N-tiles x 16 K-steps x 2 ----
#pragma unroll
        for (int t = 0; t < 8; ++t) {
            v8f acc2[2];
            const float bv = sB2[t * 16 + nl];
#pragma unroll
            for (int r = 0; r < 8; ++r) { acc2[0][r] = bv; acc2[1][r] = bv; }
#pragma unroll
            for (int kk = 0; kk < 16; ++kk) {
                const v2f bf =
                    *(const v2f*)&sW2t[(t * 16 + nl) * W2S + kk * 4 + h2];
                acc2[0] = __builtin_amdgcn_wmma_f32_16x16x4_f32(
                    false, a2[0][kk], false, bf, (short)0, acc2[0], false, false);
                acc2[1] = __builtin_amdgcn_wmma_f32_16x16x4_f32(
                    false, a2[1][kk], false, bf, (short)0, acc2[1], false, false);
            }
            // ReLU + segment max. Non-negative floats order like u32 bits,
            // so global_atomic_max_u32 on the bit pattern is exact.
#pragma unroll
            for (int u = 0; u < 2; ++u)
#pragma unroll
                for (int r = 0; r < 8; ++r) {
                    const float v = fmaxf(acc2[u][r], 0.0f);
                    atomicMax(out + (long)sidx[u][r] * EMB + t * 16 + nl,
                              __float_as_uint(v));
                }
        }
    }
}

extern "C" void kernel_launch(void* const* d_in, const int* in_sizes, int n_in,
                              void* d_out, int out_size, void* d_ws, size_t ws_size,
                              hipStream_t stream) {
    using namespace spcfg;
    const float* X  = (const float*)d_in[0];
    const int*   sg = (const int*)d_in[1];
    const float* W1 = (const float*)d_in[2];
    const float* b1 = (const float*)d_in[3];
    const float* W2 = (const float*)d_in[4];
    const float* b2 = (const float*)d_in[5];

    // zero-init output (valid float 0.0f bit pattern for the u32 atomic max)
    int n4 = out_size / 4;
    Superpoint_MAE_zero_kernel<<<(n4 + 255) / 256, 256, 0, stream>>>(
        (float4*)d_out, n4);

    Superpoint_MAE_mlp_segmax_kernel<<<GRID, BLOCK, 0, stream>>>(
        X, sg, W1, b1, W2, b2, (unsigned int*)d_out);
}